// GraphPotts_2448131358775
// MI455X (gfx1250) — compile-verified
//
#include <hip/hip_runtime.h>
#include <stdint.h>

// Problem constants (B=1, N=8192, K=32, C=21 per reference; N taken from in_sizes).
#define WAVES   8
#define TPB     (WAVES * 32)
#define CC      21
#define KK      32
#define KPW     (KK / WAVES)          // 4 tiles (k's) per wave
#define TILE_F  (CC * CC)             // 441 floats per (i,k) coupling matrix
#define SLOT_F  448                   // padded LDS slot (floats)
#define CHUNKS  14                    // ceil(441/32) async b32 issues per tile
#define TAIL_LANES (TILE_F - (CHUNKS - 1) * 32)   // 25

typedef __attribute__((address_space(1))) int as1_int;
typedef __attribute__((address_space(3))) int as3_int;

#define FENCE() asm volatile("" ::: "memory")

#if __has_builtin(__builtin_amdgcn_s_wait_asynccnt)
#define WAIT_ASYNC(n) do { __builtin_amdgcn_s_wait_asynccnt(n); FENCE(); } while (0)
#else
#define WAIT_ASYNC(n) asm volatile("s_wait_asynccnt %0" :: "i"(n) : "memory")
#endif

#define WAIT_DS0() asm volatile("s_wait_dscnt 0x0" ::: "memory")

// One async b32 copy: lane's global base + Q*128B -> lane's LDS base + Q*128B.
// The single IOFFSET applies to BOTH addresses (ISA §15.18 async pseudocode),
// so one VGPR pair (global) + one VGPR (LDS) serve all 14 chunks of a tile.
#define ASYNC_CHUNK(gp, lp, Q) \
  __builtin_amdgcn_global_load_async_to_lds_b32((gp), (lp), (Q) * 128, 0)

__device__ __forceinline__ void issue_tile(const float* g, float* l, int lane) {
  as1_int* gp = (as1_int*)(void*)(g + lane);
  as3_int* lp = (as3_int*)(void*)(l + lane);
  ASYNC_CHUNK(gp, lp, 0);
  ASYNC_CHUNK(gp, lp, 1);
  ASYNC_CHUNK(gp, lp, 2);
  ASYNC_CHUNK(gp, lp, 3);
  ASYNC_CHUNK(gp, lp, 4);
  ASYNC_CHUNK(gp, lp, 5);
  ASYNC_CHUNK(gp, lp, 6);
  ASYNC_CHUNK(gp, lp, 7);
  ASYNC_CHUNK(gp, lp, 8);
  ASYNC_CHUNK(gp, lp, 9);
  ASYNC_CHUNK(gp, lp, 10);
  ASYNC_CHUNK(gp, lp, 11);
  ASYNC_CHUNK(gp, lp, 12);
  if (lane < TAIL_LANES)                      // 416 + lane < 441
    ASYNC_CHUNK(gp, lp, 13);
}

__global__ __launch_bounds__(TPB) void potts_energy_kernel(
    const int* __restrict__ S, const float* __restrict__ h,
    const float* __restrict__ J, const int* __restrict__ edge,
    const float* __restrict__ mask_i, const float* __restrict__ mask_ij,
    float* __restrict__ out,     // d_out: [0]=U (written by reducer), [1..]=U_i
    float* __restrict__ Upart) { // d_ws:  per-node scalar contribution
  __shared__ float buf[WAVES * 2 * SLOT_F];     // double-buffered tiles, 28 KB
  __shared__ float partials[WAVES * CC];
  __shared__ float Jsh[CC];
  __shared__ float Ush[CC];

  const int i    = blockIdx.x;
  const int tid  = threadIdx.x;
  const int w    = tid >> 5;
  const int lane = tid & 31;
  const int k0   = w * KPW;

  const size_t tileBase = (size_t)i * KK;       // tile index of (i, k=0)

  // Prime the pipeline: issue tiles t=0,1 into slots 0,1.
#pragma unroll
  for (int t = 0; t < 2; ++t)
    issue_tile(J + (tileBase + (size_t)(k0 + t)) * TILE_F,
               &buf[(w * 2 + t) * SLOT_F], lane);

  float acc = 0.0f;
#pragma unroll
  for (int t = 0; t < KPW; ++t) {
    const int   k  = k0 + t;
    const int   e  = edge[i * KK + k];
    const int   sj = S[e];                       // neighbor state, in [0,C)
    const float m  = mask_ij[i * KK + k];

    if (t + 1 < KPW) WAIT_ASYNC(CHUNKS);         // tile t complete, t+1 in flight
    else             WAIT_ASYNC(0);              // last tile: drain

    const float* l = &buf[(w * 2 + (t & 1)) * SLOT_F];
    float v = 0.0f;
    if (lane < CC) v = l[lane * CC + sj];        // column gather; conflict-free
    acc = fmaf(m, v, acc);

    if (t + 2 < KPW) {                           // refill this slot
      WAIT_DS0();                                // LDS reads done before overwrite
      issue_tile(J + (tileBase + (size_t)(k0 + t + 2)) * TILE_F,
                 &buf[(w * 2 + (t & 1)) * SLOT_F], lane);
    }
  }

  if (lane < CC) partials[w * CC + lane] = acc;
  __syncthreads();

  if (tid < CC) {
    float Ji = 0.0f;
#pragma unroll
    for (int w2 = 0; w2 < WAVES; ++w2) Ji += partials[w2 * CC + tid];
    const float Uic = fmaf(h[(size_t)i * CC + tid], mask_i[i], Ji);
    out[1 + (size_t)i * CC + tid] = Uic;         // U_i
    Jsh[tid] = Ji;
    Ush[tid] = Uic;
  }
  __syncthreads();

  if (tid == 0) {
    const int si = S[i];
    Upart[i] = Ush[si] - 0.5f * Jsh[si];         // U_sel - 0.5*J_sel
  }
}

__global__ __launch_bounds__(256) void potts_reduce_kernel(
    const float* __restrict__ Upart, float* __restrict__ out, int n) {
  __shared__ float sh[256];
  float a = 0.0f;
  for (int j = threadIdx.x; j < n; j += 256) a += Upart[j];
  sh[threadIdx.x] = a;
  __syncthreads();
  for (int st = 128; st > 0; st >>= 1) {
    if ((int)threadIdx.x < st) sh[threadIdx.x] += sh[threadIdx.x + st];
    __syncthreads();
  }
  if (threadIdx.x == 0) out[0] = sh[0];          // deterministic U
}

extern "C" void kernel_launch(void* const* d_in, const int* in_sizes, int n_in,
                              void* d_out, int out_size, void* d_ws, size_t ws_size,
                              hipStream_t stream) {
  const int*   S       = (const int*)d_in[0];
  const float* h       = (const float*)d_in[1];
  const float* J       = (const float*)d_in[2];
  const int*   edge    = (const int*)d_in[3];
  const float* mask_i  = (const float*)d_in[4];
  const float* mask_ij = (const float*)d_in[5];
  float* out   = (float*)d_out;
  float* Upart = (float*)d_ws;      // N floats of scratch
  const int N = in_sizes[0];        // B*N = 8192

  potts_energy_kernel<<<N, TPB, 0, stream>>>(S, h, J, edge, mask_i, mask_ij,
                                             out, Upart);
  potts_reduce_kernel<<<1, 256, 0, stream>>>(Upart, out, N);
}